// DrugGNN_73675868996478
// MI455X (gfx1250) — compile-verified
//
#include <hip/hip_runtime.h>

// ---------------------------------------------------------------------------
// DrugGNN: segment-mean pooling (sorted segment ids) + linear embed.
//   sums[g,c]  = sum_{i: seg[i]==g} x[i,c]      (HBM-bound: 512MB read)
//   counts[g]  = #nodes in g
//   out[g,o]   = (sums[g,:]/counts[g]) . weight[o,:] + bias[o]  (f32 WMMA)
// ---------------------------------------------------------------------------

typedef __attribute__((ext_vector_type(2))) float v2f;
typedef __attribute__((ext_vector_type(8))) float v8f;

#define IN_CH   64
#define OUT_CH  128
#define CHUNK   1024   // nodes per block in the segment-sum pass

// --------------------------- zero scratch ----------------------------------
__global__ __launch_bounds__(256) void dg_zero_kernel(float* __restrict__ p, int n) {
    int i = blockIdx.x * 256 + threadIdx.x;
    int stride = gridDim.x * 256;
    for (; i < n; i += stride) p[i] = 0.0f;
}

// --------------------------- segment sum -----------------------------------
// Block = 256 threads: threadIdx.x & 15 -> float4 channel group (4 channels),
// threadIdx.x >> 4 -> row slot (16 rows in flight per iteration).
// Each thread run-length-accumulates over the sorted segment ids and flushes
// with global float atomics only at segment boundaries.
__global__ __launch_bounds__(256) void dg_segsum_kernel(
    const float4* __restrict__ x4,       // [N, 16] float4 view of x[N,64]
    const int*    __restrict__ seg,      // [N]
    float*        __restrict__ sums,     // [G, 64]
    float*        __restrict__ counts,   // [G]
    int n_nodes) {
    const int lane_c = threadIdx.x & 15;   // channel group: channels 4c..4c+3
    const int rgrp   = threadIdx.x >> 4;   // row slot 0..15
    long row = (long)blockIdx.x * CHUNK + rgrp;

    float4 acc = make_float4(0.f, 0.f, 0.f, 0.f);
    float  cnt = 0.f;
    int    cur = -1;

    #pragma unroll 4
    for (int it = 0; it < CHUNK / 16; ++it, row += 16) {
        if (row >= n_nodes) break;
        const int    s = seg[row];
        const float4 v = x4[row * (IN_CH / 4) + lane_c];
        if (s != cur) {
            if (cur >= 0) {
                float* dst = sums + (size_t)cur * IN_CH + lane_c * 4;
                atomicAdd(dst + 0, acc.x);
                atomicAdd(dst + 1, acc.y);
                atomicAdd(dst + 2, acc.z);
                atomicAdd(dst + 3, acc.w);
                if (lane_c == 0) atomicAdd(counts + cur, cnt);
            }
            cur = s;
            acc = v;
            cnt = 1.f;
        } else {
            acc.x += v.x; acc.y += v.y; acc.z += v.z; acc.w += v.w;
            cnt += 1.f;
        }
    }
    if (cur >= 0) {
        float* dst = sums + (size_t)cur * IN_CH + lane_c * 4;
        atomicAdd(dst + 0, acc.x);
        atomicAdd(dst + 1, acc.y);
        atomicAdd(dst + 2, acc.z);
        atomicAdd(dst + 3, acc.w);
        if (lane_c == 0) atomicAdd(counts + cur, cnt);
    }
}

// --------------------------- WMMA GEMM -------------------------------------
// out[G,128] = (sums/counts) @ weight^T + bias, f32 via V_WMMA_F32_16X16X4_F32.
// One wave (32 lanes) per 16x16 output tile; 4 waves per block.
// A 16x4 layout (ISA 7.12.2): lanes 0-15 hold K=k0+0/k0+1 (vgpr0/vgpr1) for
// row M=lane; lanes 16-31 hold K=k0+2/k0+3 for row M=lane-16.
// B 4x16 mirrors it: vgpr0 = row K=k0+2*half, vgpr1 = K=k0+2*half+1, N=lane&15.
// D: vgpr i -> M = i + 8*half, N = lane&15.
__global__ __launch_bounds__(128) void dg_gemm_wmma_kernel(
    const float* __restrict__ sums,     // [G, 64]
    const float* __restrict__ counts,   // [G]
    const float* __restrict__ weight,   // [128, 64] row-major
    const float* __restrict__ bias,     // [128]
    float*       __restrict__ out,      // [G, 128]
    int ntile_count) {                  // = OUT_CH/16 = 8
    const int wave = threadIdx.x >> 5;              // 0..3
    const int lane = threadIdx.x & 31;
    const int tile = blockIdx.x * 4 + wave;
    const int mt   = tile / ntile_count;            // graph-row tile
    const int nt   = tile % ntile_count;            // out-channel tile

    const int half = lane >> 4;                     // 0 | 1
    const int l16  = lane & 15;
    const int koff = half * 2;

    const int g = mt * 16 + l16;                    // A row this lane feeds
    const float rc = 1.0f / counts[g];
    const float* __restrict__ arow = sums   + (size_t)g * IN_CH;
    const float* __restrict__ brow = weight + (size_t)(nt * 16 + l16) * IN_CH;

    v8f c = {};
    #pragma unroll
    for (int k0 = 0; k0 < IN_CH; k0 += 4) {
        v2f a, b;
        a.x = arow[k0 + koff + 0] * rc;
        a.y = arow[k0 + koff + 1] * rc;
        b.x = brow[k0 + koff + 0];
        b.y = brow[k0 + koff + 1];
        // D = A(16x4,f32) * B(4x16,f32) + C  -> v_wmma_f32_16x16x4_f32
        c = __builtin_amdgcn_wmma_f32_16x16x4_f32(
                /*neg_a=*/false, a, /*neg_b=*/false, b,
                /*c_mod=*/(short)0, c, /*reuse_a=*/false, /*reuse_b=*/false);
    }

    const float bv = bias[nt * 16 + l16];
    #pragma unroll
    for (int i = 0; i < 8; ++i) {
        const int gm = mt * 16 + i + half * 8;      // output row
        out[(size_t)gm * OUT_CH + nt * 16 + l16] = c[i] + bv;
    }
}

// --------------------------- launcher --------------------------------------
extern "C" void kernel_launch(void* const* d_in, const int* in_sizes, int n_in,
                              void* d_out, int out_size, void* d_ws, size_t ws_size,
                              hipStream_t stream) {
    const float* x      = (const float*)d_in[0];   // [N, 64]
    const int*   seg    = (const int*)  d_in[1];   // [N]
    const float* weight = (const float*)d_in[2];   // [128, 64]
    const float* bias   = (const float*)d_in[3];   // [128]
    float*       out    = (float*)d_out;           // [G, 128]

    const int n_nodes = in_sizes[0] / IN_CH;       // 2,000,000
    const int G       = out_size / OUT_CH;         // 16,384

    float* sums   = (float*)d_ws;                  // G*64 floats
    float* counts = sums + (size_t)G * IN_CH;      // G floats
    const int ws_elems = G * IN_CH + G;

    // 1) zero sums + counts (harness poisons d_ws)
    {
        int blocks = (ws_elems + 255) / 256;
        if (blocks > 2048) blocks = 2048;
        dg_zero_kernel<<<blocks, 256, 0, stream>>>(sums, ws_elems);
    }

    // 2) HBM-bound segment sum with run-length register accumulation
    {
        int blocks = (n_nodes + CHUNK - 1) / CHUNK;   // 1954
        dg_segsum_kernel<<<blocks, 256, 0, stream>>>(
            (const float4*)x, seg, sums, counts, n_nodes);
    }

    // 3) f32 WMMA GEMM: (G/16)*(128/16) tiles, 4 waves/block
    {
        const int ntiles = OUT_CH / 16;               // 8
        const int tiles  = (G / 16) * ntiles;         // 8192
        dg_gemm_wmma_kernel<<<tiles / 4, 128, 0, stream>>>(
            sums, counts, weight, bias, out, ntiles);
    }
}